// GDEncoder_stu_12962211299729
// MI455X (gfx1250) — compile-verified
//
#include <hip/hip_runtime.h>
#include <hip/hip_bf16.h>

typedef float v2f __attribute__((ext_vector_type(2)));
typedef float v8f __attribute__((ext_vector_type(8)));

#define DEVINL __device__ __forceinline__

static DEVINL v8f wmma_f32_16x16x4(v2f a, v2f b, v8f c) {
  // D = A(16x4 f32) * B(4x16 f32) + C(16x16 f32), wave32
  return __builtin_amdgcn_wmma_f32_16x16x4_f32(false, a, false, b, (short)0, c, false, false);
}
static DEVINL float sigmoidf_(float x) { return 1.f / (1.f + __expf(-x)); }

constexpr int BB = 2048, KK = 12, GRID_C = 39;
constexpr int TSTU = 8, TNBR = 16;
constexpr int NN = BB * KK;                 // 24576
constexpr int EMPTY = GRID_C - KK;          // 27 empty cells share one bias-only k/v

// ---------------------------------------------------------------------------
// 1) embed: e = elu(cat(xy, cls, va) @ W1.T + b1)   rows x 5 -> rows x 32
// ---------------------------------------------------------------------------
__global__ void embed_kernel(const float* __restrict__ x2, const float* __restrict__ x1,
                             const float* __restrict__ xv, const float* __restrict__ W1,
                             const float* __restrict__ b1, float* __restrict__ e, int rows) {
  int tid = blockIdx.x * 256 + threadIdx.x;
  if (tid >= rows * 32) return;
  int r = tid >> 5, j = tid & 31;
  float f0 = x2[2 * r], f1 = x2[2 * r + 1], f2 = x1[r], f3 = xv[2 * r], f4 = xv[2 * r + 1];
  const float* w = W1 + j * 5;
  float a = b1[j] + f0 * w[0] + f1 * w[1] + f2 * w[2] + f3 * w[3] + f4 * w[4];
  e[tid] = a > 0.f ? a : expm1f(a);
}

// ---------------------------------------------------------------------------
// 2) GRU: one 16-row tile per wave, all T steps local (row-independent
//    recurrence -> no grid sync). Per output col-tile q in 0..3 the r/z/n gate
//    tiles (q, q+4, q+8) are lane-aligned with the h col-tile q, so they are
//    processed together with only ~4 v8f values live (no spills).
//    h is double-buffered in LDS (read h_cur for the K=64 GEMM, write h_next).
// ---------------------------------------------------------------------------
static DEVINL void gemm_tile(const float* __restrict__ eb, const float* __restrict__ hcur,
                             const float* __restrict__ sWih, const float* __restrict__ sWhh,
                             const float* __restrict__ sBih, const float* __restrict__ sBhh,
                             int ct, int l16, int hl, v8f& cx, v8f& ch) {
  const float bi = sBih[ct * 16 + l16];
  const float bh = sBhh[ct * 16 + l16];
  #pragma unroll
  for (int i = 0; i < 8; ++i) { cx[i] = bi; ch[i] = bh; }
  const int brow = ct * 16 + l16;
  for (int k0 = 0; k0 < 32; k0 += 4) {              // xg: e_t @ Wih.T, K=32
    int ao = l16 * 32 + k0 + 2 * hl;
    v2f a;  a[0] = eb[ao];    a[1] = eb[ao + 1];
    int bo = brow * 32 + k0 + 2 * hl;
    v2f bf; bf[0] = sWih[bo]; bf[1] = sWih[bo + 1];
    cx = wmma_f32_16x16x4(a, bf, cx);
  }
  for (int k0 = 0; k0 < 64; k0 += 4) {              // gh: h @ Whh.T, K=64
    int ao = l16 * 64 + k0 + 2 * hl;
    v2f a;  a[0] = hcur[ao];  a[1] = hcur[ao + 1];
    int bo = brow * 64 + k0 + 2 * hl;
    v2f bf; bf[0] = sWhh[bo]; bf[1] = sWhh[bo + 1];
    ch = wmma_f32_16x16x4(a, bf, ch);
  }
}

__global__ void gru_kernel(const float* __restrict__ e, const float* __restrict__ Wih,
                           const float* __restrict__ Whh, const float* __restrict__ bih,
                           const float* __restrict__ bhh, float* __restrict__ out,
                           int T, int rows, int histmode) {
  extern __shared__ float smem[];
  float* sWih = smem;             // 192*32 = 6144
  float* sWhh = smem + 6144;      // 192*64 = 12288
  float* sBih = smem + 18432;     // 192
  float* sBhh = smem + 18624;     // 192
  const int tidb = threadIdx.x;
  const int wave = tidb >> 5, lane = tidb & 31;
  const int l16 = lane & 15, hl = lane >> 4;
  float* wbase = smem + 18816 + wave * 2560;
  float* eb = wbase;              // 16x32 staged embeddings
  float* h0 = wbase + 512;        // 16x64
  float* h1 = wbase + 1536;       // 16x64

  for (int i = tidb; i < 6144; i += 128) sWih[i] = Wih[i];
  for (int i = tidb; i < 12288; i += 128) sWhh[i] = Whh[i];
  for (int i = tidb; i < 192; i += 128) { sBih[i] = bih[i]; sBhh[i] = bhh[i]; }
  __syncthreads();
  for (int i = lane; i < 1024; i += 32) h0[i] = 0.f;

  const int r0 = (blockIdx.x * 4 + wave) * 16;
  float* hc = h0;
  float* hn = h1;

  for (int t = 0; t < T; ++t) {
    const float* esrc = e + ((size_t)t * rows + r0) * 32;
    for (int i = lane; i < 512; i += 32) eb[i] = esrc[i];

    for (int q = 0; q < 4; ++q) {               // rolled: low register pressure
      v8f cx, ch, rg, zg;
      gemm_tile(eb, hc, sWih, sWhh, sBih, sBhh, q, l16, hl, cx, ch);
      #pragma unroll
      for (int i = 0; i < 8; ++i) rg[i] = sigmoidf_(cx[i] + ch[i]);
      gemm_tile(eb, hc, sWih, sWhh, sBih, sBhh, q + 4, l16, hl, cx, ch);
      #pragma unroll
      for (int i = 0; i < 8; ++i) zg[i] = sigmoidf_(cx[i] + ch[i]);
      gemm_tile(eb, hc, sWih, sWhh, sBih, sBhh, q + 8, l16, hl, cx, ch);
      #pragma unroll
      for (int i = 0; i < 8; ++i) {
        float n = tanhf(cx[i] + rg[i] * ch[i]);
        int r = i + 8 * hl, c = q * 16 + l16;
        float hold = hc[r * 64 + c];
        float hnew = (1.f - zg[i]) * n + zg[i] * hold;
        hn[r * 64 + c] = hnew;
        size_t o = histmode ? (((size_t)(r0 + r)) * T + t) * 64 + c
                            : ((size_t)t * rows + (r0 + r)) * 64 + c;
        out[o] = hnew;
      }
    }
    float* tmp = hc; hc = hn; hn = tmp;         // swap double buffers
  }
}

// ---------------------------------------------------------------------------
// 3) exchange: hid_ex[s][n][h] = sum_t Wex[s][t] * nh[t][n][h]
// ---------------------------------------------------------------------------
__global__ void exch_kernel(const float* __restrict__ nh, const float* __restrict__ Wex,
                            float* __restrict__ hid_ex) {
  const size_t plane = (size_t)NN * 64;
  size_t tid = (size_t)blockIdx.x * 256 + threadIdx.x;
  float acc[8];
  #pragma unroll
  for (int s = 0; s < 8; ++s) acc[s] = 0.f;
  for (int t = 0; t < 16; ++t) {
    float v = nh[t * plane + tid];
    #pragma unroll
    for (int s = 0; s < 8; ++s) acc[s] += Wex[s * 16 + t] * v;
  }
  #pragma unroll
  for (int s = 0; s < 8; ++s) hid_ex[s * plane + tid] = acc[s];
}

// ---------------------------------------------------------------------------
// 4) attention + GLU + LayerNorm, one wave per (b,t).
//    Attention over the grid is permutation-invariant; all 27 empty cells share
//    the bias-only k/v, realized as zero A rows (row 12) with weight 27.
// ---------------------------------------------------------------------------
__global__ void attn_kernel(const float* __restrict__ hist_hid, const float* __restrict__ hid_ex,
                            const float* __restrict__ Wq, const float* __restrict__ bq,
                            const float* __restrict__ Wk, const float* __restrict__ bk,
                            const float* __restrict__ Wv, const float* __restrict__ bv,
                            const float* __restrict__ bex,
                            const float* __restrict__ Wga, const float* __restrict__ bga,
                            const float* __restrict__ Wgg, const float* __restrict__ bgg,
                            const float* __restrict__ gamma, const float* __restrict__ beta,
                            float* __restrict__ out) {
  extern __shared__ float smem[];
  float* sWk = smem;                 // 192*64
  float* sWv = smem + 12288;         // 192*64
  float* sRk = smem + 24576;         // 192 row-sums of Wk (for bex term)
  float* sRv = smem + 24768;         // 192
  const int tidb = threadIdx.x;
  const int wave = tidb >> 5, lane = tidb & 31;
  const int l16 = lane & 15, hl = lane >> 4;
  float* wbase = smem + 24960 + wave * 7552;
  float* aex = wbase;                // 16x64 gathered neighbor hidden; reused as ctx[192]
  float* kb  = wbase + 1024;         // 16x192
  float* vb  = wbase + 4096;         // 16x192
  float* qv  = wbase + 7168;         // 192
  float* sc  = wbase + 7360;         // 64 scores/attn weights
  float* sH  = wbase + 7424;         // 64 hist row
  float* sX  = wbase + 7488;         // 64 pre-LN row

  for (int i = tidb; i < 12288; i += 128) { sWk[i] = Wk[i]; sWv[i] = Wv[i]; }
  __syncthreads();
  for (int i = tidb; i < 192; i += 128) {
    float s0 = 0.f, s1 = 0.f;
    for (int h = 0; h < 64; ++h) { s0 += sWk[i * 64 + h]; s1 += sWv[i * 64 + h]; }
    sRk[i] = s0; sRv[i] = s1;
  }
  __syncthreads();

  const int p = blockIdx.x * 4 + wave;   // flat (b,t)
  const int b = p >> 3, t = p & 7;
  const float bext = bex[t];

  for (int i = lane; i < 64; i += 32) sH[i] = hist_hid[(size_t)p * 64 + i];

  // q = hist row @ Wq.T + bq
  for (int cc = 0; cc < 6; ++cc) {
    int c = lane + 32 * cc;
    const float* wrow = Wq + (size_t)c * 64;
    float acc = bq[c];
    for (int h = 0; h < 64; ++h) acc += sH[h] * wrow[h];
    qv[c] = acc;
  }

  // gather the 12 contiguous neighbor rows for this batch; pad rows 12..15 = 0
  const float* asrc = hid_ex + ((size_t)t * NN + (size_t)b * 12) * 64;
  for (int i = lane; i < 768; i += 32) aex[i] = asrc[i];
  for (int i = 768 + lane; i < 1024; i += 32) aex[i] = 0.f;

  // K/V projection (16x192, K=64) via WMMA; zero A rows yield the empty-cell k/v
  for (int ct = 0; ct < 12; ++ct) {
    const int col = ct * 16 + l16;
    float kbias = bk[col] + bext * sRk[col];
    float vbias = bv[col] + bext * sRv[col];
    v8f ck, cv;
    #pragma unroll
    for (int i = 0; i < 8; ++i) { ck[i] = kbias; cv[i] = vbias; }
    for (int k0 = 0; k0 < 64; k0 += 4) {
      int ao = l16 * 64 + k0 + 2 * hl;
      v2f a;  a[0] = aex[ao];  a[1] = aex[ao + 1];
      int bo = col * 64 + k0 + 2 * hl;
      v2f bf; bf[0] = sWk[bo]; bf[1] = sWk[bo + 1];
      ck = wmma_f32_16x16x4(a, bf, ck);
      v2f bg; bg[0] = sWv[bo]; bg[1] = sWv[bo + 1];
      cv = wmma_f32_16x16x4(a, bg, cv);
    }
    #pragma unroll
    for (int i = 0; i < 8; ++i) {
      int r = i + 8 * hl;
      kb[r * 192 + col] = ck[i];
      vb[r * 192 + col] = cv[i];
    }
  }

  // scores: 12 neighbors + row 12 (empty cell), 4 heads, scale 1/sqrt(64)
  for (int si = lane; si < 52; si += 32) {
    int j = si >> 2, hd = si & 3;
    const float* kr = kb + j * 192 + hd * 48;
    const float* qr = qv + hd * 48;
    float acc = 0.f;
    for (int d = 0; d < 48; ++d) acc += qr[d] * kr[d];
    sc[si] = acc * 0.125f;
  }

  // softmax over 12 occupied + 27 identical empty cells (per head)
  if (lane < 4) {
    const int hd = lane;
    float m = -3.4e38f;
    #pragma unroll
    for (int j = 0; j <= 12; ++j) m = fmaxf(m, sc[j * 4 + hd]);
    float ee[13], den = 0.f;
    #pragma unroll
    for (int j = 0; j <= 12; ++j) ee[j] = __expf(sc[j * 4 + hd] - m);
    #pragma unroll
    for (int j = 0; j < 12; ++j) den += ee[j];
    den += (float)EMPTY * ee[12];
    float inv = 1.f / den;
    #pragma unroll
    for (int j = 0; j < 12; ++j) sc[j * 4 + hd] = ee[j] * inv;
    sc[52 + hd] = (float)EMPTY * ee[12] * inv;
  }

  // ctx (192) into aex[0..191]
  float ctxv[6];
  #pragma unroll
  for (int cc = 0; cc < 6; ++cc) {
    int c = lane + 32 * cc;
    int hd = c / 48;
    float acc = sc[52 + hd] * vb[12 * 192 + c];
    #pragma unroll
    for (int j = 0; j < 12; ++j) acc += sc[j * 4 + hd] * vb[j * 192 + c];
    ctxv[cc] = acc;
  }
  #pragma unroll
  for (int cc = 0; cc < 6; ++cc) aex[lane + 32 * cc] = ctxv[cc];

  // GLU + residual (Wga/Wgg are L2-resident; 96KB total)
  for (int u = 0; u < 2; ++u) {
    int h = lane * 2 + u;
    const float* wa = Wga + (size_t)h * 192;
    const float* wg = Wgg + (size_t)h * 192;
    float ga = bga[h], gg = bgg[h];
    for (int c = 0; c < 192; ++c) { float cx = aex[c]; ga += cx * wa[c]; gg += cx * wg[c]; }
    sX[h] = sH[h] + ga * sigmoidf_(gg);
  }
  // LayerNorm over 64 (redundant per-lane reduction via LDS broadcast reads)
  float mu = 0.f;
  for (int h = 0; h < 64; ++h) mu += sX[h];
  mu *= (1.f / 64.f);
  float var = 0.f;
  for (int h = 0; h < 64; ++h) { float d = sX[h] - mu; var += d * d; }
  var *= (1.f / 64.f);
  float inv = rsqrtf(var + 1e-5f);
  for (int u = 0; u < 2; ++u) {
    int h = lane * 2 + u;
    out[(size_t)p * 64 + h] = (sX[h] - mu) * inv * gamma[h] + beta[h];
  }
}

// ---------------------------------------------------------------------------
extern "C" void kernel_launch(void* const* d_in, const int* in_sizes, int n_in,
                              void* d_out, int out_size, void* d_ws, size_t ws_size,
                              hipStream_t stream) {
  const float* hist    = (const float*)d_in[0];
  const float* cls     = (const float*)d_in[1];
  const float* va      = (const float*)d_in[2];
  /* d_in[3] lane: unused (F=5) */
  const float* nbrs    = (const float*)d_in[4];
  const float* nbrscls = (const float*)d_in[5];
  const float* nbrsva  = (const float*)d_in[6];
  /* d_in[7] nbrslane: unused; d_in[8] cell_idx: unused (grid-permutation invariance) */
  const float* W1  = (const float*)d_in[9];
  const float* b1  = (const float*)d_in[10];
  const float* Wih = (const float*)d_in[11];
  const float* Whh = (const float*)d_in[12];
  const float* bih = (const float*)d_in[13];
  const float* bhh = (const float*)d_in[14];
  const float* Wq  = (const float*)d_in[15];
  const float* bq  = (const float*)d_in[16];
  const float* Wk  = (const float*)d_in[17];
  const float* bk  = (const float*)d_in[18];
  const float* Wv  = (const float*)d_in[19];
  const float* bv  = (const float*)d_in[20];
  const float* Wex = (const float*)d_in[21];
  const float* bex = (const float*)d_in[22];
  const float* Wga = (const float*)d_in[23];
  const float* bga = (const float*)d_in[24];
  const float* Wgg = (const float*)d_in[25];
  const float* bgg = (const float*)d_in[26];
  const float* gam = (const float*)d_in[27];
  const float* bet = (const float*)d_in[28];
  float* out = (float*)d_out;

  float* ws       = (float*)d_ws;
  float* e_hist   = ws;                     // 8*2048*32      =    524288
  float* e_nbrs   = e_hist + 524288;        // 16*24576*32    =  12582912
  float* hist_hid = e_nbrs + 12582912;      // 2048*8*64      =   1048576
  float* nh       = hist_hid + 1048576;     // 16*24576*64    =  25165824
  float* hid_ex   = nh + 25165824;          // 8*24576*64     =  12582912

  embed_kernel<<<2048, 256, 0, stream>>>(hist, cls, va, W1, b1, e_hist, TSTU * BB);
  embed_kernel<<<49152, 256, 0, stream>>>(nbrs, nbrscls, nbrsva, W1, b1, e_nbrs, TNBR * NN);
  // shared: (18816 + 4*2560) * 4B = 116224B
  gru_kernel<<<32, 128, 116224, stream>>>(e_hist, Wih, Whh, bih, bhh, hist_hid, TSTU, BB, 1);
  gru_kernel<<<384, 128, 116224, stream>>>(e_nbrs, Wih, Whh, bih, bhh, nh, TNBR, NN, 0);
  exch_kernel<<<6144, 256, 0, stream>>>(nh, Wex, hid_ex);
  attn_kernel<<<4096, 128, 220672, stream>>>(hist_hid, hid_ex, Wq, bq, Wk, bk, Wv, bv,
                                             bex, Wga, bga, Wgg, bgg, gam, bet, out);
}